// MultiHeadGATLayer_66288525246885
// MI455X (gfx1250) — compile-verified
//
#include <hip/hip_runtime.h>
#include <math.h>

#define N_NODES 50000
#define N_EDGES 1600000
#define IN_CH 128
#define EDGE_DIM 16
#define H 4
#define C 32
#define HC 128
#define NEG_SLOPE 0.2f
#define LN_EPS 1e-5f

typedef float v2f __attribute__((ext_vector_type(2)));
typedef float v8f __attribute__((ext_vector_type(8)));

// ---------------------------------------------------------------- utilities
__device__ __forceinline__ float atomic_max_float(float* addr, float value) {
    // sign-aware int atomic trick; correct for all IEEE floats given -FLT_MAX init
    if (value >= 0.0f) {
        return __int_as_float(atomicMax((int*)addr, __float_as_int(value)));
    } else {
        return __uint_as_float(atomicMin((unsigned int*)addr, __float_as_uint(value)));
    }
}

// ---------------------------------------------------------------- init
__global__ void gat_init_kernel(float* __restrict__ acc, float* __restrict__ amax,
                                float* __restrict__ denom, int n_acc, int n_nh) {
    int i = blockIdx.x * blockDim.x + threadIdx.x;
    if (i < n_acc) acc[i] = 0.0f;
    if (i < n_nh) {
        amax[i] = -3.402823466e38f;
        denom[i] = 0.0f;
    }
}

// ---------------------------------------------------------------- fold att_edge into W_edge
// wcomb[h*16+k] = sum_c att_edge[h,c] * W_edge[(h*32+c)*16 + k]
__global__ void gat_wecomb_kernel(const float* __restrict__ W_edge,
                                  const float* __restrict__ att_edge,
                                  float* __restrict__ wcomb) {
    int t = threadIdx.x;           // 64 threads: h = t>>4, k = t&15
    int h = t >> 4;
    int k = t & 15;
    float s = 0.0f;
#pragma unroll
    for (int c = 0; c < C; ++c)
        s += att_edge[h * C + c] * W_edge[(h * C + c) * EDGE_DIM + k];
    wcomb[t] = s;
}

// ---------------------------------------------------------------- node GEMM via WMMA f32
// xl[m, n] = sum_k x[m,k] * W_lin[n,k]   (xl = x @ W_lin^T), 16x16 tile per wave
__global__ void gat_gemm_xl_kernel(const float* __restrict__ x,
                                   const float* __restrict__ Wlin,
                                   float* __restrict__ xl) {
    const int lane = threadIdx.x;          // wave32
    const int m0 = blockIdx.x * 16;        // node tile
    const int n0 = blockIdx.y * 16;        // out-channel tile
    const int row = lane & 15;
    const int khalf = lane >> 4;           // 0: K={k0,k0+1}  1: K={k0+2,k0+3}

    v8f c = {};
    const float* arow = x + (size_t)(m0 + row) * IN_CH;     // A: row m0+row
    const float* brow = Wlin + (size_t)(n0 + row) * IN_CH;  // B[k][n] = Wlin[n*128+k]

#pragma unroll 4
    for (int k0 = 0; k0 < IN_CH; k0 += 4) {
        int kb = k0 + 2 * khalf;
        v2f a, b;
        a.x = arow[kb + 0];
        a.y = arow[kb + 1];
        b.x = brow[kb + 0];
        b.y = brow[kb + 1];
        c = __builtin_amdgcn_wmma_f32_16x16x4_f32(
                /*neg_a=*/false, a, /*neg_b=*/false, b,
                /*c_mod=*/(short)0, c, /*reuse_a=*/false, /*reuse_b=*/false);
    }

    // C/D layout: VGPR v: lanes 0-15 -> M=v, lanes 16-31 -> M=v+8; N = lane&15
    const int ncol = n0 + row;
#pragma unroll
    for (int v = 0; v < 8; ++v) {
        int mrow = m0 + v + 8 * khalf;
        xl[(size_t)mrow * HC + ncol] = c[v];
    }
}

// ---------------------------------------------------------------- per-node attention scalars
// a_src[n,h] = dot(xl[n,h,:], att_src[h,:]) ; a_dst likewise
__global__ void gat_node_score_kernel(const float* __restrict__ xl,
                                      const float* __restrict__ att_src,
                                      const float* __restrict__ att_dst,
                                      float* __restrict__ a_src,
                                      float* __restrict__ a_dst, int NH) {
    int i = blockIdx.x * blockDim.x + threadIdx.x;
    if (i >= NH) return;
    int h = i & (H - 1);
    int n = i >> 2;
    const float* rowp = xl + (size_t)n * HC + h * C;
    const float* as = att_src + h * C;
    const float* ad = att_dst + h * C;
    float ss = 0.0f, sd = 0.0f;
#pragma unroll
    for (int cc = 0; cc < C; ++cc) {
        float v = rowp[cc];
        ss += v * as[cc];
        sd += v * ad[cc];
    }
    a_src[i] = ss;
    a_dst[i] = sd;
}

// ---------------------------------------------------------------- edge pass 1: raw alpha + segment max
__global__ void gat_edge_alpha_kernel(const int* __restrict__ ei,
                                      const float* __restrict__ ea,
                                      const float* __restrict__ wcomb,
                                      const float* __restrict__ a_src,
                                      const float* __restrict__ a_dst,
                                      float* __restrict__ alpha,
                                      float* __restrict__ amax, int E) {
    __shared__ float wc[H * EDGE_DIM];
    if (threadIdx.x < H * EDGE_DIM) wc[threadIdx.x] = wcomb[threadIdx.x];
    __syncthreads();

    int e = blockIdx.x * blockDim.x + threadIdx.x;
    if (e >= E) return;
    int s = ei[e];
    int d = ei[E + e];

    const float4* eap = (const float4*)(ea + (size_t)e * EDGE_DIM);
    float4 e0 = eap[0], e1 = eap[1], e2 = eap[2], e3 = eap[3];
    float eav[EDGE_DIM] = {e0.x, e0.y, e0.z, e0.w, e1.x, e1.y, e1.z, e1.w,
                           e2.x, e2.y, e2.z, e2.w, e3.x, e3.y, e3.z, e3.w};

#pragma unroll
    for (int h = 0; h < H; ++h) {
        float dotv = 0.0f;
#pragma unroll
        for (int k = 0; k < EDGE_DIM; ++k) dotv += eav[k] * wc[h * EDGE_DIM + k];
        // NOTE: reference applies att_src to x_i (=dst) and att_dst to x_j (=src)
        float al = a_src[d * H + h] + a_dst[s * H + h] + dotv;
        al = (al > 0.0f) ? al : NEG_SLOPE * al;   // leaky relu
        alpha[(size_t)e * H + h] = al;
        atomic_max_float(&amax[d * H + h], al);
    }
}

// ---------------------------------------------------------------- edge pass 2: exp + segment sum
__global__ void gat_edge_exp_kernel(const int* __restrict__ ei,
                                    float* __restrict__ alpha,
                                    const float* __restrict__ amax,
                                    float* __restrict__ denom, int E) {
    int e = blockIdx.x * blockDim.x + threadIdx.x;
    if (e >= E) return;
    int d = ei[E + e];
#pragma unroll
    for (int h = 0; h < H; ++h) {
        float v = __expf(alpha[(size_t)e * H + h] - amax[d * H + h]);
        alpha[(size_t)e * H + h] = v;
        atomicAdd(&denom[d * H + h], v);
    }
}

// ---------------------------------------------------------------- edge pass 3: weighted scatter
// one wave32 per edge; lane covers channel h*32+lane for each head h
__global__ void gat_edge_scatter_kernel(const int* __restrict__ ei,
                                        const float* __restrict__ alpha,
                                        const float* __restrict__ denom,
                                        const float* __restrict__ xl,
                                        float* __restrict__ acc, int E) {
    int lane = threadIdx.x & 31;
    int e = (blockIdx.x * blockDim.x + threadIdx.x) >> 5;
    if (e >= E) return;
    int s = ei[e];
    int d = ei[E + e];

    float wv = 0.0f;
    if (lane < H) wv = alpha[(size_t)e * H + lane] / (denom[d * H + lane] + 1e-16f);

    const float* srow = xl + (size_t)s * HC;
    float* drow = acc + (size_t)d * HC;
#pragma unroll
    for (int h = 0; h < H; ++h) {
        float w = __shfl(wv, h, 32);
        int ch = h * C + lane;
        atomicAdd(&drow[ch], srow[ch] * w);
    }
}

// ---------------------------------------------------------------- finalize: bias+residual+LN+ELU
// one wave32 per node, 4 channels per lane, shuffle reduction (wave32 masks)
__global__ void gat_finalize_kernel(const float* __restrict__ x,
                                    const float* __restrict__ bias,
                                    const float* __restrict__ gamma,
                                    const float* __restrict__ beta,
                                    float* __restrict__ out, int N) {
    int lane = threadIdx.x & 31;
    int n = (blockIdx.x * blockDim.x + threadIdx.x) >> 5;
    if (n >= N) return;

    int ch0 = lane * 4;
    float4 o = *(const float4*)(out + (size_t)n * HC + ch0);
    float4 xr = *(const float4*)(x + (size_t)n * HC + ch0);
    float4 bb = *(const float4*)(bias + ch0);
    float v[4] = {o.x + xr.x + bb.x, o.y + xr.y + bb.y,
                  o.z + xr.z + bb.z, o.w + xr.w + bb.w};

    float s = v[0] + v[1] + v[2] + v[3];
#pragma unroll
    for (int off = 16; off > 0; off >>= 1) s += __shfl_xor(s, off, 32);
    float mu = s * (1.0f / HC);

    float q = 0.0f;
#pragma unroll
    for (int j = 0; j < 4; ++j) { float dlt = v[j] - mu; q += dlt * dlt; }
#pragma unroll
    for (int off = 16; off > 0; off >>= 1) q += __shfl_xor(q, off, 32);
    float inv = rsqrtf(q * (1.0f / HC) + LN_EPS);

    float4 g = *(const float4*)(gamma + ch0);
    float4 be = *(const float4*)(beta + ch0);
    float gv[4] = {g.x, g.y, g.z, g.w};
    float bv[4] = {be.x, be.y, be.z, be.w};
    float r[4];
#pragma unroll
    for (int j = 0; j < 4; ++j) {
        float y = (v[j] - mu) * inv * gv[j] + bv[j];
        r[j] = (y > 0.0f) ? y : (__expf(y) - 1.0f);   // ELU
    }
    *(float4*)(out + (size_t)n * HC + ch0) = make_float4(r[0], r[1], r[2], r[3]);
}

// ---------------------------------------------------------------- launch
extern "C" void kernel_launch(void* const* d_in, const int* in_sizes, int n_in,
                              void* d_out, int out_size, void* d_ws, size_t ws_size,
                              hipStream_t stream) {
    const float* x        = (const float*)d_in[0];
    const int*   ei       = (const int*)d_in[1];
    const float* ea       = (const float*)d_in[2];
    const float* W_lin    = (const float*)d_in[3];
    const float* W_edge   = (const float*)d_in[4];
    const float* att_src  = (const float*)d_in[5];
    const float* att_dst  = (const float*)d_in[6];
    const float* att_edge = (const float*)d_in[7];
    const float* bias     = (const float*)d_in[8];
    const float* ln_gamma = (const float*)d_in[9];
    const float* ln_beta  = (const float*)d_in[10];
    float* out = (float*)d_out;

    // workspace layout (floats)
    float* ws = (float*)d_ws;
    size_t off = 0;
    float* xl    = ws + off; off += (size_t)N_NODES * HC;   // 6.4M
    float* a_src = ws + off; off += (size_t)N_NODES * H;    // 0.2M
    float* a_dst = ws + off; off += (size_t)N_NODES * H;
    float* amax  = ws + off; off += (size_t)N_NODES * H;
    float* denom = ws + off; off += (size_t)N_NODES * H;
    float* alpha = ws + off; off += (size_t)N_EDGES * H;    // 6.4M
    float* wcomb = ws + off; off += H * EDGE_DIM;

    const int n_acc = N_NODES * HC;
    const int n_nh  = N_NODES * H;

    // 1) init accumulator (d_out) + amax/denom
    gat_init_kernel<<<(n_acc + 255) / 256, 256, 0, stream>>>(out, amax, denom, n_acc, n_nh);

    // 2) fold att_edge into W_edge
    gat_wecomb_kernel<<<1, 64, 0, stream>>>(W_edge, att_edge, wcomb);

    // 3) node GEMM via WMMA f32 16x16x4 (one 16x16 tile per wave)
    gat_gemm_xl_kernel<<<dim3(N_NODES / 16, HC / 16), 32, 0, stream>>>(x, W_lin, xl);

    // 4) per-node attention scalars
    gat_node_score_kernel<<<(n_nh + 255) / 256, 256, 0, stream>>>(xl, att_src, att_dst,
                                                                  a_src, a_dst, n_nh);

    // 5) edge passes: alpha + segmax, exp + segsum, weighted scatter
    gat_edge_alpha_kernel<<<(N_EDGES + 255) / 256, 256, 0, stream>>>(ei, ea, wcomb, a_src,
                                                                     a_dst, alpha, amax, N_EDGES);
    gat_edge_exp_kernel<<<(N_EDGES + 255) / 256, 256, 0, stream>>>(ei, alpha, amax, denom, N_EDGES);
    gat_edge_scatter_kernel<<<(N_EDGES * 32 + 255) / 256, 256, 0, stream>>>(ei, alpha, denom,
                                                                            xl, out, N_EDGES);

    // 6) finalize: bias + residual + layernorm + ELU (in-place on d_out)
    gat_finalize_kernel<<<(N_NODES * 32 + 255) / 256, 256, 0, stream>>>(x, bias, ln_gamma,
                                                                        ln_beta, out, N_NODES);
}